// CostomAdaptiveAvgPool2D_47802986004510
// MI455X (gfx1250) — compile-verified
//
#include <hip/hip_runtime.h>

typedef __attribute__((ext_vector_type(2))) float v2f;
typedef __attribute__((ext_vector_type(8))) float v8f;

#define H_IN 64
#define W_IN 64
#define ROW_STRIDE 68                       // padded LDS row stride (floats): 16B aligned, bank-conflict free
#define IMG_FLOATS (H_IN * ROW_STRIDE)      // 4352 floats / image
#define IMGS_PER_BLOCK 2
#define PW_STRIDE 68                        // padded PwT row stride (floats)
#define Y_ROW 8                             // padded Y row stride (floats)
#define BLOCK 256

// Pw[w][j]: 1/wlen_j if w falls in adaptive bin j (64 -> 6), else 0.
// starts = floor(j*64/6), ends = ceil((j+1)*64/6); lens are 11 or 12.
__device__ __forceinline__ float pool_w_weight(int w, int j) {
  if (j >= 6) return 0.0f;                  // padded N columns
  int ws = (j * 64) / 6;
  int we = ((j + 1) * 64 + 5) / 6;
  float iw = ((we - ws) == 12) ? (1.0f / 12.0f) : (1.0f / 11.0f);
  return (w >= ws && w < we) ? iw : 0.0f;
}

__global__ __launch_bounds__(BLOCK) void adaptive_avgpool_wmma(
    const float* __restrict__ x, float* __restrict__ out) {
  __shared__ float lds_x[IMGS_PER_BLOCK * IMG_FLOATS];          // 34,816 B
  __shared__ float lds_pw[16 * PW_STRIDE];                      //  4,352 B
  __shared__ float lds_y[IMGS_PER_BLOCK * H_IN * Y_ROW];        //  4,096 B

  const int t = threadIdx.x;
  const int lane = t & 31;
  const int wave = t >> 5;

  // ---------- stage 0: coalesced global -> LDS (2 images as float4) ----------
  const float4* gx =
      (const float4*)(x + (size_t)blockIdx.x * IMGS_PER_BLOCK * H_IN * W_IN);
  #pragma unroll
  for (int it = 0; it < 8; ++it) {
    int idx4 = t + it * BLOCK;              // 0..2047 float4s
    int img  = idx4 >> 10;                  // 1024 float4s per image
    int rem  = idx4 & 1023;
    int row  = rem >> 4;                    // 16 float4s per row
    int c4   = rem & 15;
    float4 v = gx[idx4];
    *(float4*)&lds_x[img * IMG_FLOATS + row * ROW_STRIDE + c4 * 4] = v;
  }

  // ---------- build PwT[j][w] (transposed pooling weights) once per block ----------
  #pragma unroll
  for (int it = 0; it < 4; ++it) {
    int idx = t * 4 + it;                   // 0..1023
    int j = idx >> 6;
    int w = idx & 63;
    lds_pw[j * PW_STRIDE + w] = pool_w_weight(w, j);
  }

  __syncthreads();

  // ---------- stage 1: Y = X * Pw via V_WMMA_F32_16X16X4_F32 ----------
  // 8 waves: wave -> (image = wave/4, 16-row tile = wave%4)
  const int jn  = lane & 15;
  const int kh  = lane >> 4;
  const int img     = wave >> 2;
  const int rowbase = (wave & 3) * 16;

  // B fragments: lanes 0-15 -> K={k0,k0+1} col jn; lanes 16-31 -> K={k0+2,k0+3}
  const float* pws = &lds_pw[jn * PW_STRIDE + 2 * kh];
  v2f bf[16];
  #pragma unroll
  for (int kk = 0; kk < 16; ++kk)
    bf[kk] = *(const v2f*)(pws + kk * 4);   // ds_load_b64, conflict-free

  // A fragments: lanes 0-15 row=jn K0/K1; lanes 16-31 row=jn K2/K3
  const float* xs = &lds_x[img * IMG_FLOATS + (rowbase + jn) * ROW_STRIDE + 2 * kh];
  v2f af[16];
  #pragma unroll
  for (int kk = 0; kk < 16; ++kk)
    af[kk] = *(const v2f*)(xs + kk * 4);    // ds_load_b64, conflict-free

  v8f acc = {};
  #pragma unroll
  for (int kk = 0; kk < 16; ++kk)
    acc = __builtin_amdgcn_wmma_f32_16x16x4_f32(
        /*neg_a=*/false, af[kk], /*neg_b=*/false, bf[kk],
        /*c_mod=*/(short)0, acc, /*reuse_a=*/false, /*reuse_b=*/false);

  // D layout: VGPR v -> row (rowbase + v + 8*kh), col = jn. Keep cols 0..5.
  if (jn < 6) {
    #pragma unroll
    for (int v = 0; v < 8; ++v) {
      int row = rowbase + v + 8 * kh;
      lds_y[img * (H_IN * Y_ROW) + row * Y_ROW + jn] = acc[v];
    }
  }

  __syncthreads();

  // ---------- stage 2: out[i][j] = (1/hlen_i) * sum_{h in bin i} Y[h][j] ----------
  if (t < IMGS_PER_BLOCK * 36) {
    int im = t / 36;
    int r  = t % 36;
    int i  = r / 6;
    int j  = r % 6;
    int hs = (i * 64) / 6;
    int he = ((i + 1) * 64 + 5) / 6;
    float ih = ((he - hs) == 12) ? (1.0f / 12.0f) : (1.0f / 11.0f);
    float s = 0.0f;
    for (int h = hs; h < he; ++h)
      s += lds_y[im * (H_IN * Y_ROW) + h * Y_ROW + j];
    out[((size_t)blockIdx.x * IMGS_PER_BLOCK + im) * 36 + r] = s * ih;
  }
}

extern "C" void kernel_launch(void* const* d_in, const int* in_sizes, int n_in,
                              void* d_out, int out_size, void* d_ws, size_t ws_size,
                              hipStream_t stream) {
  const float* x = (const float*)d_in[0];
  float* out = (float*)d_out;
  int n_imgs = in_sizes[0] / (H_IN * W_IN);       // 16*2048 = 32768
  int blocks = n_imgs / IMGS_PER_BLOCK;           // 16384
  adaptive_avgpool_wmma<<<blocks, BLOCK, 0, stream>>>(x, out);
}